// QLayer_65481071399248
// MI455X (gfx1250) — compile-verified
//
#include <hip/hip_runtime.h>
#include <math.h>
#include <stdint.h>

typedef float v2f __attribute__((ext_vector_type(2)));
typedef float v8f __attribute__((ext_vector_type(8)));

#define NQ      10
#define DIM     1024            // 2^10 state dimension
#define B_TOTAL 16384
#define ZSIZE   (B_TOTAL * NQ)  // 163840 output elements
#define NCHUNK  8

// ws layout (floats):
//   [0, 160)                      : 20 fused gates, 8 floats each
//   [256, 256+DIM*DIM)            : WPr  (Re W^T, k-paired layout [k/2][j][k&1])
//   [256+DIM*DIM, 256+2*DIM*DIM)  : WPi  (Im W^T, same layout)
//   [256+2*DIM*DIM, +8*ZSIZE)     : per-chunk z partials
#define WR_OFF   256
#define WI_OFF   (256 + DIM * DIM)
#define PART_OFF (256 + 2 * DIM * DIM)

static __device__ __forceinline__ void cmulf(float ar, float ai, float br, float bi,
                                             float& cr, float& ci) {
  cr = ar * br - ai * bi;
  ci = ar * bi + ai * br;
}

// ------------- K1: fuse U = RY(p2) * RX(p1) * RZ(p0) per (layer,qubit) -------------
__global__ void k1_build_gates(const float* __restrict__ params, float* __restrict__ G) {
  int t = threadIdx.x;
  if (t >= 20) return;
  float a = params[t * 3 + 0], b = params[t * 3 + 1], c = params[t * 3 + 2];
  float cz = cosf(0.5f * a), sz = sinf(0.5f * a);
  float cb = cosf(0.5f * b), sb = sinf(0.5f * b);
  float cc = cosf(0.5f * c), sc = sinf(0.5f * c);
  // ez = e^{-ia/2} = (cz, -sz)
  float u00r, u00i, u01r, u01i, u10r, u10i, u11r, u11i;
  cmulf(cz, -sz,  cc * cb,  sc * sb, u00r, u00i);
  cmulf(cz,  sz, -sc * cb, -cc * sb, u01r, u01i);
  cmulf(cz, -sz,  sc * cb, -cc * sb, u10r, u10i);
  cmulf(cz,  sz,  cc * cb, -sc * sb, u11r, u11i);
  float* g = G + t * 8;
  g[0] = u00r; g[1] = u00i; g[2] = u01r; g[3] = u01i;
  g[4] = u10r; g[5] = u10i; g[6] = u11r; g[7] = u11i;
}

// ------------- K2: build the fixed unitary W (column per basis state) -------------
// One wave32 per basis state k. Amplitude index j = r*32 + lane.
// Output layout is WMMA-fragment-ready: WP[(k>>1)*2048 + j*2 + (k&1)].
__global__ void k2_build_unitary(const float* __restrict__ G,
                                 float* __restrict__ WPr, float* __restrict__ WPi) {
  int lane = threadIdx.x & 31;
  int k = (blockIdx.x << 3) | (threadIdx.x >> 5);  // 0..1023

  float sr[32], si[32];
  #pragma unroll
  for (int r = 0; r < 32; ++r) {
    sr[r] = ((r << 5) + lane == k) ? 1.0f : 0.0f;
    si[r] = 0.0f;
  }

  #pragma unroll
  for (int l = 0; l < 2; ++l) {
    // ---- 10 fused 1-qubit gates ----
    #pragma unroll
    for (int q = 0; q < NQ; ++q) {
      const float* g = G + (l * 10 + q) * 8;
      float u00r = g[0], u00i = g[1], u01r = g[2], u01i = g[3];
      float u10r = g[4], u10i = g[5], u11r = g[6], u11i = g[7];
      if (q < 5) {
        const int m = 1 << (4 - q);
        #pragma unroll
        for (int r = 0; r < 32; ++r) {
          if (r & m) continue;
          int r1 = r | m;
          float a0r = sr[r], a0i = si[r], a1r = sr[r1], a1i = si[r1];
          sr[r]  = u00r * a0r - u00i * a0i + u01r * a1r - u01i * a1i;
          si[r]  = u00r * a0i + u00i * a0r + u01r * a1i + u01i * a1r;
          sr[r1] = u10r * a0r - u10i * a0i + u11r * a1r - u11i * a1i;
          si[r1] = u10r * a0i + u10i * a0r + u11r * a1i + u11i * a1r;
        }
      } else {
        const int lm = 1 << (9 - q);
        bool up = (lane & lm) != 0;
        float c0r = up ? u10r : u00r, c0i = up ? u10i : u00i;
        float c1r = up ? u11r : u01r, c1i = up ? u11i : u01i;
        #pragma unroll
        for (int r = 0; r < 32; ++r) {
          float pr = __shfl_xor(sr[r], lm, 32);
          float pi = __shfl_xor(si[r], lm, 32);
          float a0r = up ? pr : sr[r], a0i = up ? pi : si[r];
          float a1r = up ? sr[r] : pr, a1i = up ? si[r] : pi;
          sr[r] = c0r * a0r - c0i * a0i + c1r * a1r - c1i * a1i;
          si[r] = c0r * a0i + c0i * a0r + c1r * a1i + c1i * a1r;
        }
      }
    }
    // ---- CNOT chain ----
    #pragma unroll
    for (int c = 0; c < 9; ++c) {
      const int pc = 9 - c, pt = 8 - c;
      if (pc >= 6) {
        const int cm = 1 << (pc - 5), tm = 1 << (pt - 5);
        #pragma unroll
        for (int r = 0; r < 32; ++r) {
          if ((r & cm) && !(r & tm)) {
            int r1 = r | tm;
            float t;
            t = sr[r]; sr[r] = sr[r1]; sr[r1] = t;
            t = si[r]; si[r] = si[r1]; si[r1] = t;
          }
        }
      } else if (pc == 5) {
        #pragma unroll
        for (int r = 0; r < 32; ++r) {
          if (r & 1) {
            sr[r] = __shfl_xor(sr[r], 16, 32);
            si[r] = __shfl_xor(si[r], 16, 32);
          }
        }
      } else {
        const int cm = 1 << pc, tm = 1 << pt;
        bool ctl = (lane & cm) != 0;
        #pragma unroll
        for (int r = 0; r < 32; ++r) {
          float tr = __shfl_xor(sr[r], tm, 32);
          float ti = __shfl_xor(si[r], tm, 32);
          sr[r] = ctl ? tr : sr[r];
          si[r] = ctl ? ti : si[r];
        }
      }
    }
    // ---- wrap CNOT(9,0) ----
    {
      bool ctl = (lane & 1) != 0;
      #pragma unroll
      for (int r = 0; r < 16; ++r) {
        int r1 = r + 16;
        float n0r = ctl ? sr[r1] : sr[r];
        float n0i = ctl ? si[r1] : si[r];
        float n1r = ctl ? sr[r]  : sr[r1];
        float n1i = ctl ? si[r]  : si[r1];
        sr[r] = n0r; si[r] = n0i; sr[r1] = n1r; si[r1] = n1i;
      }
    }
  }

  // k-paired store: WP[(k>>1)*2048 + j*2 + (k&1)]
  float* wr = WPr + (size_t)(k >> 1) * 2048 + (k & 1);
  float* wi = WPi + (size_t)(k >> 1) * 2048 + (k & 1);
  #pragma unroll
  for (int r = 0; r < 32; ++r) {
    wr[(size_t)((r << 5) + lane) * 2] = sr[r];
    wi[(size_t)((r << 5) + lane) * 2] = si[r];
  }
}

// ------------- K3: LDS-staged f32 WMMA GEMM + Z-expectation epilogue -------------
// Block = 8 waves = 128 batch rows; all waves share W tiles staged to LDS with
// async loads (double-buffered, ASYNCcnt). Wave tile: M=16 j-rows x N=16 batch,
// K accumulated 1024 in 8 stages of 128 via V_WMMA_F32_16X16X4_F32.
__global__ __launch_bounds__(256) void k3_gemm_z(const float* __restrict__ x,
                                                 const float* __restrict__ WPr,
                                                 const float* __restrict__ WPi,
                                                 float* __restrict__ PART) {
  __shared__ __align__(16) float bufR[2][2048];  // [buf][k2(64) * 32 floats]
  __shared__ __align__(16) float bufI[2][2048];

  const int tid   = threadIdx.x;
  const int lane  = tid & 31;
  const int wv    = tid >> 5;
  const int blk   = blockIdx.x;   // 0..1023
  const int chunk = blk & 7;      // 8 j-tiles per chunk
  const int bt8   = blk >> 3;     // 0..127: 128-row batch super-tile
  const int bcol  = lane & 15;
  const int half  = lane >> 4;
  const int b     = bt8 * 128 + wv * 16 + bcol;

  // per-sample half-angle cos/sin (lanes 16..31 mirror 0..15 -> same b)
  float hc[NQ], hs[NQ];
  const float* xb = x + b * NQ;
  #pragma unroll
  for (int q = 0; q < NQ; ++q) {
    float h = 0.5f * xb[q];
    hc[q] = cosf(h);
    hs[q] = sinf(h);
  }
  // psi0 low-bit products, pre-selected for this lane half (k bits 0..4 = qubits 9..5)
  float fsel[16];
  {
    float lo[32];
    #pragma unroll
    for (int t = 0; t < 32; ++t) {
      float v = (t & 1)  ? hs[9] : hc[9];
      v *= (t & 2)  ? hs[8] : hc[8];
      v *= (t & 4)  ? hs[7] : hc[7];
      v *= (t & 8)  ? hs[6] : hc[6];
      v *= (t & 16) ? hs[5] : hc[5];
      lo[t] = v;
    }
    #pragma unroll
    for (int t = 0; t < 8; ++t) {
      fsel[2 * t]     = half ? lo[4 * t + 2] : lo[4 * t + 0];
      fsel[2 * t + 1] = half ? lo[4 * t + 3] : lo[4 * t + 1];
    }
  }

  float zacc[NQ];
  #pragma unroll
  for (int i = 0; i < NQ; ++i) zacc[i] = 0.0f;

  const int k2r = tid >> 3;  // staging row 0..31 (+32 for rep 1)
  const int o16 = tid & 7;   // 16-byte piece within a 128B row

  for (int jj = 0; jj < 8; ++jj) {
    const int j0 = ((chunk << 3) + jj) << 4;
    v8f cre = {0.f, 0.f, 0.f, 0.f, 0.f, 0.f, 0.f, 0.f};
    v8f cim = {0.f, 0.f, 0.f, 0.f, 0.f, 0.f, 0.f, 0.f};

    // ---- issue stage 0 into buffer 0 (4 async B128 loads per thread) ----
    #pragma unroll
    for (int rep = 0; rep < 2; ++rep) {
      const int row = k2r + rep * 32;
      const size_t gfo = (size_t)row * 2048 + (size_t)j0 * 2 + o16 * 4;
      uint32_t lR = (uint32_t)(uintptr_t)&bufR[0][row * 32 + o16 * 4];
      uint32_t lI = (uint32_t)(uintptr_t)&bufI[0][row * 32 + o16 * 4];
      uint64_t aR = (uint64_t)(uintptr_t)(WPr + gfo);
      uint64_t aI = (uint64_t)(uintptr_t)(WPi + gfo);
      asm volatile("global_load_async_to_lds_b128 %0, %1, off" :: "v"(lR), "v"(aR) : "memory");
      asm volatile("global_load_async_to_lds_b128 %0, %1, off" :: "v"(lI), "v"(aI) : "memory");
    }

    for (int s = 0; s < 8; ++s) {
      if (s + 1 < 8) {
        // prefetch next 128-K stage into the other buffer
        const int d = (s + 1) & 1;
        #pragma unroll
        for (int rep = 0; rep < 2; ++rep) {
          const int row = k2r + rep * 32;
          const size_t gfo = (size_t)((s + 1) * 64 + row) * 2048 + (size_t)j0 * 2 + o16 * 4;
          uint32_t lR = (uint32_t)(uintptr_t)&bufR[d][row * 32 + o16 * 4];
          uint32_t lI = (uint32_t)(uintptr_t)&bufI[d][row * 32 + o16 * 4];
          uint64_t aR = (uint64_t)(uintptr_t)(WPr + gfo);
          uint64_t aI = (uint64_t)(uintptr_t)(WPi + gfo);
          asm volatile("global_load_async_to_lds_b128 %0, %1, off" :: "v"(lR), "v"(aR) : "memory");
          asm volatile("global_load_async_to_lds_b128 %0, %1, off" :: "v"(lI), "v"(aI) : "memory");
        }
        asm volatile("s_wait_asynccnt 0x4" ::: "memory");  // current stage complete
      } else {
        asm volatile("s_wait_asynccnt 0x0" ::: "memory");
      }
      __syncthreads();

      const int cur = s & 1;
      // stage-uniform part of psi0 hi factor (k bits 7..9 -> qubits 2..0)
      float hvs = (s & 1) ? hs[2] : hc[2];
      hvs *= (s & 2) ? hs[1] : hc[1];
      hvs *= (s & 4) ? hs[0] : hc[0];

      #pragma unroll
      for (int h2 = 0; h2 < 4; ++h2) {
        // k bits 5,6 -> qubits 4,3
        float hv = hvs * ((h2 & 1) ? hs[4] : hc[4]) * ((h2 & 2) ? hs[3] : hc[3]);
        #pragma unroll
        for (int kl = 0; kl < 32; kl += 4) {
          const int k2l = h2 * 16 + (kl >> 1) + half;
          v2f a_re, a_im, bf;
          a_re.x = bufR[cur][k2l * 32 + bcol * 2 + 0];
          a_re.y = bufR[cur][k2l * 32 + bcol * 2 + 1];
          a_im.x = bufI[cur][k2l * 32 + bcol * 2 + 0];
          a_im.y = bufI[cur][k2l * 32 + bcol * 2 + 1];
          bf.x = hv * fsel[(kl >> 1) + 0];
          bf.y = hv * fsel[(kl >> 1) + 1];
          cre = __builtin_amdgcn_wmma_f32_16x16x4_f32(false, a_re, false, bf,
                                                      (short)0, cre, false, false);
          cim = __builtin_amdgcn_wmma_f32_16x16x4_f32(false, a_im, false, bf,
                                                      (short)0, cim, false, false);
        }
      }
      __syncthreads();  // all waves done reading buf[cur] before it is refilled
    }

    // epilogue: prob = re^2+im^2, signed accumulation per qubit
    // C/D layout: VGPR v -> row M = v + 8*half, col N = bcol
    #pragma unroll
    for (int v = 0; v < 8; ++v) {
      float p = cre[v] * cre[v] + cim[v] * cim[v];
      int j = j0 + v + (half << 3);
      #pragma unroll
      for (int i = 0; i < NQ; ++i)
        zacc[i] += ((j >> (9 - i)) & 1) ? -p : p;
    }
  }

  // combine the two lane halves (same b, disjoint j rows), write chunk partial
  #pragma unroll
  for (int i = 0; i < NQ; ++i) zacc[i] += __shfl_xor(zacc[i], 16, 32);
  if (lane < 16) {
    float* dst = PART + (size_t)chunk * ZSIZE + (size_t)b * NQ;
    #pragma unroll
    for (int i = 0; i < NQ; ++i) dst[i] = zacc[i];
  }
}

// ------------- K4: deterministic combine of the 8 chunk partials -------------
__global__ void k4_combine(const float* __restrict__ PART, float* __restrict__ out) {
  int t = blockIdx.x * blockDim.x + threadIdx.x;
  if (t >= ZSIZE) return;
  float s = 0.0f;
  #pragma unroll
  for (int c = 0; c < NCHUNK; ++c) s += PART[(size_t)c * ZSIZE + t];
  out[t] = s;
}

extern "C" void kernel_launch(void* const* d_in, const int* in_sizes, int n_in,
                              void* d_out, int out_size, void* d_ws, size_t ws_size,
                              hipStream_t stream) {
  const float* x      = (const float*)d_in[0];  // [16384, 10] f32
  const float* params = (const float*)d_in[1];  // [2, 10, 3] f32
  float* out = (float*)d_out;                   // [16384, 10] f32
  float* ws  = (float*)d_ws;                    // needs ~13.6 MB

  float* G    = ws;             // 160 floats
  float* WPr  = ws + WR_OFF;    // 1024*1024 (k-paired)
  float* WPi  = ws + WI_OFF;    // 1024*1024 (k-paired)
  float* PART = ws + PART_OFF;  // 8 * 163840

  k1_build_gates<<<1, 32, 0, stream>>>(params, G);
  k2_build_unitary<<<128, 256, 0, stream>>>(G, WPr, WPi);
  k3_gemm_z<<<1024, 256, 0, stream>>>(x, WPr, WPi, PART);
  k4_combine<<<(ZSIZE + 255) / 256, 256, 0, stream>>>(PART, out);
}